// GlobalModel_2714419331676
// MI455X (gfx1250) — compile-verified
//
#include <hip/hip_runtime.h>
#include <hip/hip_bf16.h>

typedef __attribute__((ext_vector_type(2))) float v2f;
typedef __attribute__((ext_vector_type(4))) float v4f;
typedef __attribute__((ext_vector_type(8))) float v8f;

#define D_NODE 128
#define D_GLOB 128
#define HIDDEN 512
#define D_OUT  128
#define SEG_CHUNK 256   // nodes per block (4 waves x 64 nodes)

// ---------------------------------------------------------------------------
// Kernel 0: zero scratch (sums + counts)
// ---------------------------------------------------------------------------
__global__ void zero_kernel(float* __restrict__ p, int n) {
    int i = blockIdx.x * blockDim.x + threadIdx.x;
    if (i < n) p[i] = 0.0f;
}

// ---------------------------------------------------------------------------
// Kernel 1: segment-sum of x[N,128] by sorted batch[] into sums[B,128], cnts[B].
// 128-thread block = 4 waves; each wave owns 64 contiguous nodes of the chunk.
// Lane l owns features 4l..4l+3  ->  one non-temporal global_load_b128 per row
// (512B coalesced per wave-instruction; x is stream-once so NT keeps L2 for
// the weights/atomic lines). batch is sorted -> register accumulation with a
// wave-uniform atomic flush on segment change (~2.5MB total atomic traffic).
// ---------------------------------------------------------------------------
__global__ void __launch_bounds__(128)
segsum_kernel(const float* __restrict__ x, const long long* __restrict__ batch,
              float* __restrict__ sums, float* __restrict__ cnts, int N) {
    __shared__ long long sb[SEG_CHUNK];
    const int start = blockIdx.x * SEG_CHUNK;
    if (start >= N) return;
    int end = start + SEG_CHUNK;
    if (end > N) end = N;
    const int cnt = end - start;

    for (int i = threadIdx.x; i < cnt; i += 128) sb[i] = batch[start + i];
    __syncthreads();

    const int wv   = threadIdx.x >> 5;          // wave 0..3
    const int lane = threadIdx.x & 31;
    const int c4   = lane * 4;                  // feature base for this lane

    int wbeg = wv * (SEG_CHUNK / 4);
    int wend = wbeg + (SEG_CHUNK / 4);
    if (wend > cnt) wend = cnt;
    if (wbeg >= cnt) return;                    // no barriers after this point

    v4f acc = {0.f, 0.f, 0.f, 0.f};
    int cur = (int)sb[wbeg];
    int run = 0;
    for (int i = wbeg; i < wend; ++i) {
        int b = (int)sb[i];
        if (b != cur) {                         // uniform within the wave
            float* s = &sums[cur * D_NODE + c4];
            atomicAdd(s + 0, acc.x);
            atomicAdd(s + 1, acc.y);
            atomicAdd(s + 2, acc.z);
            atomicAdd(s + 3, acc.w);
            if (lane == 0) atomicAdd(&cnts[cur], (float)run);
            acc = (v4f){0.f, 0.f, 0.f, 0.f};
            run = 0;
            cur = b;
        }
        const v4f* px = (const v4f*)(x + (size_t)(start + i) * D_NODE + c4);
        v4f v = __builtin_nontemporal_load(px); // global_load_b128, TH=NT
        acc += v;
        ++run;
    }
    float* s = &sums[cur * D_NODE + c4];
    atomicAdd(s + 0, acc.x);
    atomicAdd(s + 1, acc.y);
    atomicAdd(s + 2, acc.z);
    atomicAdd(s + 3, acc.w);
    if (lane == 0) atomicAdd(&cnts[cur], (float)run);
}

// ---------------------------------------------------------------------------
// Kernel 2: fused  out = relu([u | mean] @ W1 + b1) @ W2 + b2  via fp32 WMMA.
// One block (4 waves, 128 thr) per 16-row tile of the batch.
//   Phase 1: wave w computes H[:, w*128 .. w*128+127] as 8 WMMA acc tiles,
//            K-loop over 256 in steps of 4 (v_wmma_f32_16x16x4_f32).
//   Phase 2: bias+ReLU applied in the C/D fragment layout, spilled to LDS.
//   Phase 3: wave w computes output N-tiles {2w, 2w+1} over full K=512.
// ---------------------------------------------------------------------------
__global__ void __launch_bounds__(128)
mlp_kernel(const float* __restrict__ u,
           const float* __restrict__ sums, const float* __restrict__ cnts,
           const float* __restrict__ W1, const float* __restrict__ b1,
           const float* __restrict__ W2, const float* __restrict__ b2,
           float* __restrict__ out) {
    __shared__ float lds_A[16][D_GLOB + D_NODE + 1]; // 16 x 257 (pad: conflict-free col reads)
    __shared__ float lds_H[16][HIDDEN + 1];          // 16 x 513

    const int tid = threadIdx.x;
    const int m0  = blockIdx.x * 16;

    // Build A tile = [u | sums/cnt] for rows m0..m0+15 (cooperative, coalesced)
    for (int i = tid; i < 16 * (D_GLOB + D_NODE); i += 128) {
        int r = i >> 8;          // row in tile
        int c = i & 255;         // column 0..255
        int row = m0 + r;
        float v;
        if (c < D_GLOB) {
            v = u[row * D_GLOB + c];
        } else {
            v = sums[row * D_NODE + (c - D_GLOB)] / fmaxf(cnts[row], 1.0f);
        }
        lds_A[r][c] = v;
    }
    __syncthreads();

    const int lane  = tid & 31;
    const int wv    = tid >> 5;       // wave id 0..3
    const int aRow  = lane & 15;      // A-fragment: M = lane&15
    const int hi    = lane >> 4;      //             K half select
    const int nlane = lane & 15;      // B/C/D-fragment: N = lane&15

    const v8f vzero = {0.f, 0.f, 0.f, 0.f, 0.f, 0.f, 0.f, 0.f};

    // ---- GEMM 1: H[16, 512] = A[16,256] @ W1[256,512], wave owns 128 cols ----
    v8f acc[8];
    #pragma unroll
    for (int t = 0; t < 8; ++t) acc[t] = vzero;

    for (int k0 = 0; k0 < D_GLOB + D_NODE; k0 += 4) {
        // A frag (16x4 f32): lane L<16 -> {K=k0,k0+1}, lane L+16 -> {K=k0+2,k0+3}
        v2f a;
        a.x = lds_A[aRow][k0 + 2 * hi];
        a.y = lds_A[aRow][k0 + 2 * hi + 1];
        #pragma unroll
        for (int t = 0; t < 8; ++t) {
            int n0 = wv * 128 + t * 16;
            // B frag (4x16 f32), mirrored layout: row-major W1, coalesced 16-lane reads
            v2f bf;
            bf.x = W1[(k0 + 2 * hi)     * HIDDEN + n0 + nlane];
            bf.y = W1[(k0 + 2 * hi + 1) * HIDDEN + n0 + nlane];
            acc[t] = __builtin_amdgcn_wmma_f32_16x16x4_f32(
                false, a, false, bf, (short)0, acc[t], false, false);
        }
    }

    // bias + ReLU in D-fragment layout; spill H to LDS (row M = r + 8*hi, col N)
    #pragma unroll
    for (int t = 0; t < 8; ++t) {
        int n0 = wv * 128 + t * 16;
        float bv = b1[n0 + nlane];
        #pragma unroll
        for (int r = 0; r < 8; ++r) {
            float h = acc[t][r] + bv;
            lds_H[r + 8 * hi][n0 + nlane] = fmaxf(h, 0.0f);
        }
    }
    __syncthreads();

    // ---- GEMM 2: out[16,128] = H[16,512] @ W2[512,128]; wave owns 2 N-tiles ----
    v8f acc2[2];
    acc2[0] = vzero;
    acc2[1] = vzero;
    for (int k0 = 0; k0 < HIDDEN; k0 += 4) {
        v2f a;
        a.x = lds_H[aRow][k0 + 2 * hi];
        a.y = lds_H[aRow][k0 + 2 * hi + 1];
        #pragma unroll
        for (int j = 0; j < 2; ++j) {
            int n0 = (2 * wv + j) * 16;
            v2f bf;
            bf.x = W2[(k0 + 2 * hi)     * D_OUT + n0 + nlane];
            bf.y = W2[(k0 + 2 * hi + 1) * D_OUT + n0 + nlane];
            acc2[j] = __builtin_amdgcn_wmma_f32_16x16x4_f32(
                false, a, false, bf, (short)0, acc2[j], false, false);
        }
    }

    // bias + store (D layout: VGPR r, lanes<16 -> M=r, lanes>=16 -> M=r+8)
    #pragma unroll
    for (int j = 0; j < 2; ++j) {
        int n = (2 * wv + j) * 16 + nlane;
        float bv = b2[n];
        #pragma unroll
        for (int r = 0; r < 8; ++r) {
            int m = m0 + r + 8 * hi;
            out[m * D_OUT + n] = acc2[j][r] + bv;
        }
    }
}

// ---------------------------------------------------------------------------
// Launch: zero scratch -> bandwidth-bound segment sum -> fused WMMA MLP
// Inputs (setup_inputs order): x, edge_index, edge_attr, u, batch, W1, b1, W2, b2
// ---------------------------------------------------------------------------
extern "C" void kernel_launch(void* const* d_in, const int* in_sizes, int n_in,
                              void* d_out, int out_size, void* d_ws, size_t ws_size,
                              hipStream_t stream) {
    const float*     x     = (const float*)d_in[0];
    const float*     u     = (const float*)d_in[3];
    const long long* batch = (const long long*)d_in[4];
    const float*     W1    = (const float*)d_in[5];
    const float*     b1    = (const float*)d_in[6];
    const float*     W2    = (const float*)d_in[7];
    const float*     b2    = (const float*)d_in[8];
    float*           out   = (float*)d_out;

    const int N = in_sizes[0] / D_NODE;   // 1,000,000
    const int B = in_sizes[3] / D_GLOB;   // 1024

    float* sums = (float*)d_ws;              // [B * 128]
    float* cnts = sums + (size_t)B * D_NODE; // [B]

    const int zn = B * D_NODE + B;
    zero_kernel<<<(zn + 255) / 256, 256, 0, stream>>>(sums, zn);

    const int nblk = (N + SEG_CHUNK - 1) / SEG_CHUNK;
    segsum_kernel<<<nblk, 128, 0, stream>>>(x, batch, sums, cnts, N);

    mlp_kernel<<<B / 16, 128, 0, stream>>>(u, sums, cnts, W1, b1, W2, b2, out);
}